// MultiScaleLocalCon_33174327394661
// MI455X (gfx1250) — compile-verified
//
#include <hip/hip_runtime.h>
#include <hip/hip_bf16.h>

// ---------------------------------------------------------------------------
// MultiScaleLocalCon for MI455X (gfx1250, wave32).
//  - Gram matrix via V_WMMA_F32_16X16X4_F32 (K=4 f32 WMMA == 3-D dot products)
//  - dist + top-33 fused in one workgroup using a 168 KB LDS slab (CDNA5 WGP
//    LDS is 320 KB) -> removes the 16 MB HBM round-trip of the naive version
//  - points staged into LDS with global_load_async_to_lds (ASYNCcnt path)
// ---------------------------------------------------------------------------

typedef __attribute__((ext_vector_type(2))) float v2f;
typedef __attribute__((ext_vector_type(8))) float v8f;
typedef int msl_i4 __attribute__((vector_size(16)));   // builtin's b128 pointee

#define NPTS   8192
#define MAXU   2048
#define NBK    33          // 32+1 neighbors (max scale); smaller scales are prefixes

// ---- async global->LDS staging (gfx1250), with safe fallbacks -------------
#if defined(__has_builtin)
#  if __has_builtin(__builtin_amdgcn_global_load_async_to_lds_b128)
#    define MSL_ASYNC_B128 1
#  endif
#  if __has_builtin(__builtin_amdgcn_global_load_async_to_lds_b32)
#    define MSL_ASYNC_B32 1
#  endif
#  if __has_builtin(__builtin_amdgcn_s_wait_asynccnt)
#    define MSL_WAIT_ASYNC() __builtin_amdgcn_s_wait_asynccnt(0)
#  endif
#endif
#if !defined(MSL_WAIT_ASYNC)
#  define MSL_WAIT_ASYNC() asm volatile("s_wait_asynccnt 0x0" ::: "memory")
#endif

#define MSL_AS1 __attribute__((address_space(1)))
#define MSL_AS3 __attribute__((address_space(3)))
// typed, address-space-qualified casts matching the builtin signatures
#define MSL_G128(p) ((MSL_AS1 msl_i4*)(void*)(p))
#define MSL_L128(p) ((MSL_AS3 msl_i4*)(void*)(p))
#define MSL_G32(p)  ((MSL_AS1 int*)(void*)(p))
#define MSL_L32(p)  ((MSL_AS3 int*)(void*)(p))

// ---------------- workspace layout (bytes, 16B-aligned chunks) -------------
static constexpr size_t OFF_CNTS   = 0;                      // 3 ints: Ur, Us, Uc
static constexpr size_t OFF_DSUM   = 256;                    // 3 floats (zeroed)
static constexpr size_t OFF_ACCC   = 272;                    // 2048 floats (zeroed)
static constexpr size_t OFF_RUNIQ  = 8704;                   // 2048 int
static constexpr size_t OFF_SUNIQ  = OFF_RUNIQ  + 8192;
static constexpr size_t OFF_KEYS   = OFF_SUNIQ  + 8192;
static constexpr size_t OFF_KUNIQ  = OFF_KEYS   + 8192;
static constexpr size_t OFF_PAIRR  = OFF_KUNIQ  + 8192;
static constexpr size_t OFF_PAIRS  = OFF_PAIRR  + 8192;
static constexpr size_t OFF_RPOS   = OFF_PAIRS  + 8192;
static constexpr size_t OFF_SPOS   = OFF_RPOS   + 8192;
static constexpr size_t OFF_SELF   = OFF_SPOS   + 8192;      // sel flags
static constexpr size_t OFF_SSCORE = OFF_SELF   + 8192;      // per-pair mean score
static constexpr size_t OFF_DSC    = OFF_SSCORE + 8192;      // 3*2048 floats dscore
static constexpr size_t OFF_SQR    = OFF_DSC    + 24576;
static constexpr size_t OFF_SQS    = OFF_SQR    + 8192;
static constexpr size_t OFF_PTSR   = OFF_SQS    + 8192;      // 2048 float4
static constexpr size_t OFF_PTSS   = OFF_PTSR   + 32768;
static constexpr size_t OFF_NIDR   = OFF_PTSS   + 32768;     // 2048*33 int
static constexpr size_t OFF_NDR    = OFF_NIDR   + 270336;
static constexpr size_t OFF_NIDS   = OFF_NDR    + 270336;
static constexpr size_t OFF_NDS    = OFF_NIDS   + 270336;

// ---------------------------------------------------------------------------
__global__ void msl_zero_f(float* __restrict__ p, int n) {
  int i = blockIdx.x * blockDim.x + threadIdx.x;
  if (i < n) p[i] = 0.0f;
}

__global__ void msl_build_keys(const int* __restrict__ r, const int* __restrict__ s,
                               int* __restrict__ keys, int n) {
  int i = blockIdx.x * blockDim.x + threadIdx.x;
  if (i < n) keys[i] = (r[i] << 13) | s[i];   // NPTS = 2^13
}

// single-workgroup bitonic sort (2048 elems in LDS) + serial unique compact
__global__ __launch_bounds__(1024) void msl_sort_unique(const int* __restrict__ in,
                                                        int n,
                                                        int* __restrict__ uout,
                                                        int* __restrict__ cnt) {
  __shared__ int s[2048];
  int tid = threadIdx.x;
  for (int i = tid; i < 2048; i += 1024) s[i] = (i < n) ? in[i] : 0x7fffffff;
  __syncthreads();
  for (int k = 2; k <= 2048; k <<= 1) {
    for (int j = k >> 1; j > 0; j >>= 1) {
      for (int i = tid; i < 2048; i += 1024) {
        int ixj = i ^ j;
        if (ixj > i) {
          int a = s[i], b = s[ixj];
          bool up = ((i & k) == 0);
          if ((a > b) == up) { s[i] = b; s[ixj] = a; }
        }
      }
      __syncthreads();
    }
  }
  if (tid == 0) {
    int c = 0;
    for (int i = 0; i < n; ++i) {
      int v = s[i];
      if (i == 0 || v != s[i - 1]) uout[c++] = v;
    }
    *cnt = c;
  }
}

__global__ void msl_decode_pairs(const int* __restrict__ keys, const int* __restrict__ cnts,
                                 int* __restrict__ pr, int* __restrict__ ps) {
  int i = blockIdx.x * blockDim.x + threadIdx.x;
  if (i < cnts[2]) { pr[i] = keys[i] >> 13; ps[i] = keys[i] & 8191; }
}

__global__ void msl_gather(const float* __restrict__ pts3, const int* __restrict__ uniq,
                           const int* __restrict__ cnt, float4* __restrict__ pts,
                           float* __restrict__ sq) {
  int i = blockIdx.x * blockDim.x + threadIdx.x;
  if (i >= MAXU) return;
  float4 p = make_float4(0.f, 0.f, 0.f, 0.f);
  float q = 0.f;
  if (i < *cnt) {
    int g = uniq[i];
    p.x = pts3[g * 3 + 0];
    p.y = pts3[g * 3 + 1];
    p.z = pts3[g * 3 + 2];
    q = p.x * p.x + p.y * p.y + p.z * p.z;
  }
  pts[i] = p;
  sq[i]  = q;
}

// ---------------------------------------------------------------------------
// Fused distances + top-33 for one 16-row slab.
// 16 waves (512 threads). LDS: 16x2048 f32 dist slab (128K) + staged points
// (32K) + norms (8K) = 168K of the WGP's 320K.
// ---------------------------------------------------------------------------
__global__ __launch_bounds__(512) void msl_dist_topk(const float4* __restrict__ pts,
                                                     const float* __restrict__ sq,
                                                     const int* __restrict__ cnt,
                                                     const int* __restrict__ uniq,
                                                     int* __restrict__ nbrId,
                                                     float* __restrict__ nbrD) {
  __shared__ float  ldist[16 * MAXU];   // 128 KB
  __shared__ float4 lpts[MAXU];         //  32 KB
  __shared__ float  lsq[MAXU];          //   8 KB
  const int U  = *cnt;
  const int tm = blockIdx.x;            // 16-row slab index
  if (tm * 16 >= U) return;             // uniform whole-block branch
  const int tid  = threadIdx.x;
  const int wave = tid >> 5;            // 0..15
  const int lane = tid & 31;

  // ---- stage all (zero-padded) unique points + norms into LDS -------------
#if defined(MSL_ASYNC_B128)
  for (int i = tid; i < MAXU; i += 512) {
    __builtin_amdgcn_global_load_async_to_lds_b128(
        MSL_G128(pts + i), MSL_L128(&lpts[i]), 0, 0);
  }
#  if defined(MSL_ASYNC_B32)
  for (int i = tid; i < MAXU; i += 512) {
    __builtin_amdgcn_global_load_async_to_lds_b32(
        MSL_G32(sq + i), MSL_L32(&lsq[i]), 0, 0);
  }
#  else
  for (int i = tid; i < MAXU; i += 512) lsq[i] = sq[i];
#  endif
  MSL_WAIT_ASYNC();
#else
  for (int i = tid; i < MAXU; i += 512) { lpts[i] = pts[i]; lsq[i] = sq[i]; }
#endif
  __syncthreads();

  // ---- WMMA: this slab's A tile vs all column tiles, dists into LDS -------
  const int  l15 = lane & 15;
  const bool hi  = lane >= 16;
  // A: 16x4 f32 (MxK) -> lanes0-15 hold K=0,1 ; lanes16-31 hold K=2,3
  float4 pa = lpts[tm * 16 + l15];
  v2f a;  a.x = hi ? pa.z : pa.x;  a.y = hi ? pa.w : pa.y;

  const int ntiles = (U + 15) >> 4;
  for (int tn = wave; tn < ntiles; tn += 16) {
    float4 pb = lpts[tn * 16 + l15];
    v2f b;  b.x = hi ? pb.z : pb.x;  b.y = hi ? pb.w : pb.y;
    v8f c = {};
    c = __builtin_amdgcn_wmma_f32_16x16x4_f32(false, a, false, b,
                                              (short)0, c, false, false);
    const int col = tn * 16 + l15;
    const float sqc = lsq[col];
#pragma unroll
    for (int v = 0; v < 8; ++v) {
      int rloc = v + (hi ? 8 : 0);
      float d2 = lsq[tm * 16 + rloc] + sqc - 2.0f * c[v];
      d2 = fmaxf(d2, 0.0f);
      ldist[rloc * MAXU + col] = (d2 > 0.0f) ? sqrtf(d2) : 0.0f;  // grad-safe sqrt
    }
  }
  __syncthreads();

  // ---- top-33 (dist,idx)-lexicographic per row, one wave per row, in LDS --
  const int row = tm * 16 + wave;
  if (row < U) {
    const float* __restrict__ drow = &ldist[wave * MAXU];
    float lastD = -1.0f;
    int   lastI = -1;
    for (int k = 0; k < NBK; ++k) {
      float bestD = 3.0e38f;
      int   bestI = 0x7fffffff;
      for (int j = lane; j < U; j += 32) {
        float d = drow[j];
        bool gt = (d > lastD) || (d == lastD && j > lastI);
        if (gt && (d < bestD || (d == bestD && j < bestI))) { bestD = d; bestI = j; }
      }
      for (int off = 16; off > 0; off >>= 1) {
        float od = __shfl_xor(bestD, off, 32);
        int   oi = __shfl_xor(bestI, off, 32);
        if (od < bestD || (od == bestD && oi < bestI)) { bestD = od; bestI = oi; }
      }
      if (lane == 0) {
        if (bestI == 0x7fffffff) { nbrId[row * NBK + k] = -1;           nbrD[row * NBK + k] = 3.0e38f; }
        else                     { nbrId[row * NBK + k] = uniq[bestI];  nbrD[row * NBK + k] = bestD; }
      }
      lastD = bestD; lastI = bestI;
    }
  }
}

__device__ __forceinline__ int msl_lower_bound(const int* __restrict__ a, int n, int v) {
  int lo = 0, hi = n;
  while (lo < hi) { int mid = (lo + hi) >> 1; if (a[mid] < v) lo = mid + 1; else hi = mid; }
  return lo;
}

__global__ void msl_positions(const int* __restrict__ pr, const int* __restrict__ ps,
                              const int* __restrict__ runiq, const int* __restrict__ suniq,
                              const int* __restrict__ cnts,
                              int* __restrict__ rpos, int* __restrict__ spos) {
  int i = blockIdx.x * blockDim.x + threadIdx.x;
  if (i < cnts[2]) {
    rpos[i] = msl_lower_bound(runiq, cnts[0], pr[i]);
    spos[i] = msl_lower_bound(suniq, cnts[1], ps[i]);
  }
}

// ---- per-scale correlation: one workgroup per unique pair j ---------------
__global__ __launch_bounds__(256) void msl_corr(const int* __restrict__ pr,
                                                const int* __restrict__ ps,
                                                const int* __restrict__ rpos,
                                                const int* __restrict__ spos,
                                                const int* __restrict__ nidR,
                                                const float* __restrict__ ndR,
                                                const int* __restrict__ nidS,
                                                const float* __restrict__ ndS,
                                                const int* __restrict__ cnts,
                                                int scale, int sc,
                                                float* __restrict__ dscore,
                                                float* __restrict__ dsum,
                                                float* __restrict__ accc) {
  const int uc = cnts[2];
  const int j = blockIdx.x;
  if (j >= uc) return;
  const int r = min(scale, cnts[0] - 1);
  const int s = min(scale, cnts[1] - 1);

  __shared__ int   rid[NBK]; __shared__ float rdd[NBK];
  __shared__ int   sid[NBK]; __shared__ float sdd[NBK];
  const int tid = threadIdx.x;
  const int rp = rpos[j], sp = spos[j];
  if (tid < NBK) {
    rid[tid] = nidR[rp * NBK + tid]; rdd[tid] = ndR[rp * NBK + tid];
    sid[tid] = nidS[sp * NBK + tid]; sdd[tid] = ndS[sp * NBK + tid];
  }
  __syncthreads();

  int cntm = 0; float acc = 0.f;
  for (int m = tid; m < uc; m += 256) {
    int rm = pr[m];
    int kr = -1;
    for (int k = 0; k <= r; ++k) if (rid[k] == rm) { kr = k; break; }
    if (kr >= 0) {
      int sm = ps[m];
      int ks = -1;
      for (int k = 0; k <= s; ++k) if (sid[k] == sm) { ks = k; break; }
      if (ks >= 0) {
        float dr = rdd[kr], dsv = sdd[ks];
        float cur = fabsf(dr - dsv) / (fmaxf(dr, dsv) + 1e-8f);
        acc += expf(-cur);
        ++cntm;
      }
    }
  }

  __shared__ float sa[256]; __shared__ int scn[256];
  sa[tid] = acc; scn[tid] = cntm;
  __syncthreads();
  for (int off = 128; off > 0; off >>= 1) {
    if (tid < off) { sa[tid] += sa[tid + off]; scn[tid] += scn[tid + off]; }
    __syncthreads();
  }
  if (tid == 0) {
    float cf = (float)scn[0];
    float dsc = (scn[0] > 0) ? sa[0] / cf : 0.f;
    dscore[sc * MAXU + j] = dsc;
    atomicAdd(&dsum[sc], dsc);
    if (cf > (float)(min(r, s) * 2)) accc[j] += 1.f;
  }
}

// ---- final selection + ordered compaction (single workgroup) --------------
__global__ __launch_bounds__(256) void msl_finalize(const int* __restrict__ cnts,
                                                    const float* __restrict__ dscore,
                                                    const float* __restrict__ dsum,
                                                    const float* __restrict__ accc,
                                                    const int* __restrict__ pr,
                                                    const int* __restrict__ ps,
                                                    int* __restrict__ self,
                                                    float* __restrict__ sscore,
                                                    float* __restrict__ out, int out_n) {
  const int uc = cnts[2];
  const int tid = threadIdx.x;
  __shared__ float mean[3];
  if (tid < 3) mean[tid] = (uc > 0) ? dsum[tid] / (float)uc : 0.f;
  __syncthreads();

  const int base_j = tid * 8;
  int loc = 0;
  for (int e = 0; e < 8; ++e) {
    int j = base_j + e;
    int flag = 0; float sc = 0.f;
    if (j < uc) {
      float d0 = dscore[j], d1 = dscore[MAXU + j], d2 = dscore[2 * MAXU + j];
      int accd = (d0 > mean[0]) + (d1 > mean[1]) + (d2 > mean[2]);
      flag = (accc[j] > 0.f) && (accd > 0);
      sc = (d0 + d1 + d2) * (1.f / 3.f);
    }
    self[j] = flag; sscore[j] = sc;
    loc += flag;
  }

  __shared__ int part[256];
  part[tid] = loc;
  __syncthreads();
  for (int off = 1; off < 256; off <<= 1) {
    int v = part[tid];
    int add = (tid >= off) ? part[tid - off] : 0;
    __syncthreads();
    part[tid] = v + add;
    __syncthreads();
  }
  const int total = part[255];
  int p = part[tid] - loc;   // exclusive prefix for this chunk
  for (int e = 0; e < 8; ++e) {
    int j = base_j + e;
    if (self[j]) {
      if (p < out_n)             out[p] = (float)pr[j];
      if (total + p < out_n)     out[total + p] = (float)ps[j];
      if (2 * total + p < out_n) out[2 * total + p] = sscore[j];
      ++p;
    }
  }
}

// ---------------------------------------------------------------------------
extern "C" void kernel_launch(void* const* d_in, const int* in_sizes, int n_in,
                              void* d_out, int out_size, void* d_ws, size_t ws_size,
                              hipStream_t stream) {
  const float* ref_point = (const float*)d_in[0];
  const float* src_point = (const float*)d_in[1];
  const int*   ref_index = (const int*)d_in[2];
  const int*   src_index = (const int*)d_in[3];
  const int n = in_sizes[2];                       // N_CORR = 2048

  char* w = (char*)d_ws;
  int*    cnts   = (int*)  (w + OFF_CNTS);
  float*  dsum   = (float*)(w + OFF_DSUM);
  float*  accc   = (float*)(w + OFF_ACCC);
  int*    runiq  = (int*)  (w + OFF_RUNIQ);
  int*    suniq  = (int*)  (w + OFF_SUNIQ);
  int*    keys   = (int*)  (w + OFF_KEYS);
  int*    kuniq  = (int*)  (w + OFF_KUNIQ);
  int*    pairr  = (int*)  (w + OFF_PAIRR);
  int*    pairs  = (int*)  (w + OFF_PAIRS);
  int*    rpos   = (int*)  (w + OFF_RPOS);
  int*    spos   = (int*)  (w + OFF_SPOS);
  int*    selfl  = (int*)  (w + OFF_SELF);
  float*  sscore = (float*)(w + OFF_SSCORE);
  float*  dscore = (float*)(w + OFF_DSC);
  float*  sqR    = (float*)(w + OFF_SQR);
  float*  sqS    = (float*)(w + OFF_SQS);
  float4* ptsR   = (float4*)(w + OFF_PTSR);
  float4* ptsS   = (float4*)(w + OFF_PTSS);
  int*    nidR   = (int*)  (w + OFF_NIDR);
  float*  ndR    = (float*)(w + OFF_NDR);
  int*    nidS   = (int*)  (w + OFF_NIDS);
  float*  ndS    = (float*)(w + OFF_NDS);

  // 0) zero accumulators (dsum+accc contiguous) and d_out (graph-replay safe)
  msl_zero_f<<<(2052 + 255) / 256, 256, 0, stream>>>(dsum, 2052);
  if (out_size > 0)
    msl_zero_f<<<(out_size + 255) / 256, 256, 0, stream>>>((float*)d_out, out_size);

  // 1) uniques (sorted) and unique pairs (lexicographic via packed keys)
  msl_build_keys<<<(n + 255) / 256, 256, 0, stream>>>(ref_index, src_index, keys, n);
  msl_sort_unique<<<1, 1024, 0, stream>>>(ref_index, n, runiq, &cnts[0]);
  msl_sort_unique<<<1, 1024, 0, stream>>>(src_index, n, suniq, &cnts[1]);
  msl_sort_unique<<<1, 1024, 0, stream>>>(keys,      n, kuniq, &cnts[2]);
  msl_decode_pairs<<<(MAXU + 255) / 256, 256, 0, stream>>>(kuniq, cnts, pairr, pairs);

  // 2) gather unique points (zero-padded to 2048 so WMMA tiles never diverge)
  msl_gather<<<(MAXU + 255) / 256, 256, 0, stream>>>(ref_point, runiq, &cnts[0], ptsR, sqR);
  msl_gather<<<(MAXU + 255) / 256, 256, 0, stream>>>(src_point, suniq, &cnts[1], ptsS, sqS);

  // 3) fused WMMA distances + top-33 per row, all in LDS (ref side, src side)
  msl_dist_topk<<<MAXU / 16, 512, 0, stream>>>(ptsR, sqR, &cnts[0], runiq, nidR, ndR);
  msl_dist_topk<<<MAXU / 16, 512, 0, stream>>>(ptsS, sqS, &cnts[1], suniq, nidS, ndS);

  // 4) searchsorted positions of each pair endpoint in the unique tables
  msl_positions<<<(MAXU + 255) / 256, 256, 0, stream>>>(pairr, pairs, runiq, suniq,
                                                        cnts, rpos, spos);

  // 5) per-scale correlation (scales 8, 16, 32)
  const int co[3] = {8, 16, 32};
  for (int sc = 0; sc < 3; ++sc)
    msl_corr<<<MAXU, 256, 0, stream>>>(pairr, pairs, rpos, spos,
                                       nidR, ndR, nidS, ndS,
                                       cnts, co[sc], sc, dscore, dsum, accc);

  // 6) selection + ordered compaction into d_out: [ref_ids | src_ids | scores]
  msl_finalize<<<1, 256, 0, stream>>>(cnts, dscore, dsum, accc, pairr, pairs,
                                      selfl, sscore, (float*)d_out, out_size);
}